// HybridQLSTMTagger_65481071402283
// MI455X (gfx1250) — compile-verified
//
#include <hip/hip_runtime.h>
#include <math.h>

#define SEQ   8192
#define TAGS  128
#define EMB   2

typedef __attribute__((ext_vector_type(2))) float v2f;
typedef __attribute__((ext_vector_type(8))) float v8f;

// ---------------------------------------------------------------------------
// complex helpers (float2-style)
// ---------------------------------------------------------------------------
struct c2 { float x, y; };

// ---------------------------------------------------------------------------
// Kernel 1: photonic_extract — 2-qubit (4 amp) circuit per sequence position
// state index = w0*2 + w1  (wire0 = bit mask 2, wire1 = bit mask 1)
// ---------------------------------------------------------------------------
struct FraudP {
  float bs_theta, bs_phi;
  float ph0, ph1;
  float sq_r0, sq_r1, sq_phi0, sq_phi1;
  float d_r0, d_r1, d_phi0, d_phi1;
  float k0, k1;
};
// clip(|r|<=5, |kerr|<=1) is a no-op for these constants
constexpr FraudP LAY_IN = { 0.7f,  0.3f,  0.5f, -0.4f,  0.6f,  0.8f,  0.2f, -0.1f,
                            0.4f,  0.5f,  0.1f,  0.3f,  0.2f, -0.3f };
constexpr FraudP LAY_0  = { 0.9f, -0.2f,  0.3f,  0.6f,  0.5f, -0.7f,  0.4f,  0.2f,
                            0.6f, -0.3f, -0.2f,  0.5f,  0.4f,  0.1f };
constexpr FraudP LAY_1  = {-0.5f,  0.8f, -0.6f,  0.2f,  0.3f,  0.9f, -0.3f,  0.5f,
                            0.2f,  0.7f,  0.6f, -0.4f, -0.5f,  0.3f };

__device__ __forceinline__ void rx2(c2 psi[4], float half_t, int mask) {
  float s, c;
  __sincosf(half_t, &s, &c);
#pragma unroll
  for (int i0 = 0; i0 < 4; ++i0) {
    if (i0 & mask) continue;
    const int i1 = i0 | mask;
    c2 a0 = psi[i0], a1 = psi[i1];
    // out0 = c*a0 + (-i s)*a1 ; out1 = (-i s)*a0 + c*a1
    psi[i0] = { c * a0.x + s * a1.y,  c * a0.y - s * a1.x };
    psi[i1] = { s * a0.y + c * a1.x, -s * a0.x + c * a1.y };
  }
}

__device__ __forceinline__ void bs_apply(c2 psi[4], float ct, float st, float cp, float sp) {
  // out1 = ct*psi1 - e^{i phi} st * psi2 ; out2 = e^{-i phi} st * psi1 + ct * psi2
  c2 a1 = psi[1], a2 = psi[2];
  // e^{i phi} * a2
  c2 t1 = { cp * a2.x - sp * a2.y, cp * a2.y + sp * a2.x };
  // e^{-i phi} * a1
  c2 t2 = { cp * a1.x + sp * a1.y, cp * a1.y - sp * a1.x };
  psi[1] = { ct * a1.x - st * t1.x, ct * a1.y - st * t1.y };
  psi[2] = { st * t2.x + ct * a2.x, st * t2.y + ct * a2.y };
}

__device__ __forceinline__ void rot2w(c2 psi[4], float r, float phi, int mask) {
  float sr, cr, sp, cp;
  __sincosf(r, &sr, &cr);
  __sincosf(phi, &sp, &cp);
  const c2 m01 = { -cp * sr, -sp * sr };   // -e^{i phi} sin r
  const c2 m10 = {  cp * sr, -sp * sr };   //  e^{-i phi} sin r
#pragma unroll
  for (int i0 = 0; i0 < 4; ++i0) {
    if (i0 & mask) continue;
    const int i1 = i0 | mask;
    c2 a0 = psi[i0], a1 = psi[i1];
    psi[i0] = { cr * a0.x + m01.x * a1.x - m01.y * a1.y,
                cr * a0.y + m01.x * a1.y + m01.y * a1.x };
    psi[i1] = { m10.x * a0.x - m10.y * a0.y + cr * a1.x,
                m10.x * a0.y + m10.y * a0.x + cr * a1.y };
  }
}

__device__ __forceinline__ void kerrw(c2 psi[4], float k, int mask) {
  float sk, ck;
  __sincosf(k, &sk, &ck);
#pragma unroll
  for (int i = 0; i < 4; ++i) {
    if (i & mask) {
      c2 a = psi[i];
      psi[i] = { ck * a.x - sk * a.y, ck * a.y + sk * a.x };
    }
  }
}

__device__ __forceinline__ void fraud_layer(c2 psi[4], const FraudP p) {
  float st, ct, sp, cp;
  __sincosf(p.bs_theta, &st, &ct);
  __sincosf(p.bs_phi, &sp, &cp);
  bs_apply(psi, ct, st, cp, sp);
  rx2(psi, 0.5f * p.ph0, 2);
  rx2(psi, 0.5f * p.ph1, 1);
  rot2w(psi, p.sq_r0, p.sq_phi0, 2);
  rot2w(psi, p.sq_r1, p.sq_phi1, 1);
  bs_apply(psi, ct, st, cp, sp);
  rx2(psi, 0.5f * p.ph0, 2);
  rx2(psi, 0.5f * p.ph1, 1);
  rot2w(psi, p.d_r0, p.d_phi0, 2);
  rot2w(psi, p.d_r1, p.d_phi1, 1);
  kerrw(psi, p.k0, 2);
  kerrw(psi, p.k1, 1);
}

__global__ void __launch_bounds__(256)
photonic_kernel(const float* __restrict__ emb, const int* __restrict__ sentence,
                float* __restrict__ feats) {
  const int s = blockIdx.x * blockDim.x + threadIdx.x;
  if (s >= SEQ) return;
  const int tok = sentence[s];
  const float x0 = emb[tok * EMB + 0];
  const float x1 = emb[tok * EMB + 1];
  c2 psi[4] = { {1.f, 0.f}, {0.f, 0.f}, {0.f, 0.f}, {0.f, 0.f} };
  rx2(psi, 0.5f * x0, 2);
  rx2(psi, 0.5f * x1, 1);
  fraud_layer(psi, LAY_IN);
  fraud_layer(psi, LAY_0);
  fraud_layer(psi, LAY_1);
  const float p0 = psi[0].x * psi[0].x + psi[0].y * psi[0].y;
  const float p1 = psi[1].x * psi[1].x + psi[1].y * psi[1].y;
  const float p2 = psi[2].x * psi[2].x + psi[2].y * psi[2].y;
  const float p3 = psi[3].x * psi[3].x + psi[3].y * psi[3].y;
  feats[s * 2 + 0] = p0 + p1 - p2 - p3;   // Z on wire 0 (bit 2)
  feats[s * 2 + 1] = p0 - p1 + p2 - p3;   // Z on wire 1 (bit 1)
}

// ---------------------------------------------------------------------------
// Kernel 2: qLSTM scan — single wave32, lane computes gate (lane & 3)
// 4-wire state: index = w0*8 + w1*4 + w2*2 + w3 ; wire w mask = 8 >> w
// Consecutive RX on the same wire commute: RX(pre[w]) RX(theta[w]) = RX(sum)
// ---------------------------------------------------------------------------
__device__ __forceinline__ void rx16(float ax[16], float ay[16], float half_t, int mask) {
  float s, c;
  __sincosf(half_t, &s, &c);
#pragma unroll
  for (int i0 = 0; i0 < 16; ++i0) {
    if (i0 & mask) continue;
    const int i1 = i0 | mask;
    const float a0x = ax[i0], a0y = ay[i0], a1x = ax[i1], a1y = ay[i1];
    ax[i0] =  c * a0x + s * a1y;  ay[i0] =  c * a0y - s * a1x;
    ax[i1] =  s * a0y + c * a1x;  ay[i1] = -s * a0x + c * a1y;
  }
}

__device__ __forceinline__ void cnot16(float ax[16], float ay[16], int cm, int tm) {
#pragma unroll
  for (int i = 0; i < 16; ++i) {
    if ((i & cm) && !(i & tm)) {
      const int j = i | tm;
      float tx = ax[i], ty = ay[i];
      ax[i] = ax[j]; ay[i] = ay[j];
      ax[j] = tx;    ay[j] = ty;
    }
  }
}

__device__ __forceinline__ float sigmoidf(float x) {
  return 1.0f / (1.0f + __expf(-x));
}

__global__ void __launch_bounds__(32)
qlstm_kernel(const float* __restrict__ feats, const float* __restrict__ W,
             const float* __restrict__ b, const float* __restrict__ thetas,
             float* __restrict__ hs) {
  const int lane = threadIdx.x & 31;
  const int g = lane & 3;          // gate handled by this lane (replicated x8)

  float Wg[24];
#pragma unroll
  for (int i = 0; i < 24; ++i) Wg[i] = W[g * 24 + i];
  float bg[4], th[4];
#pragma unroll
  for (int o = 0; o < 4; ++o) { bg[o] = b[g * 4 + o]; th[o] = thetas[g * 4 + o]; }

  float hx[4] = {0.f, 0.f, 0.f, 0.f};
  float cx[4] = {0.f, 0.f, 0.f, 0.f};

  for (int s = 0; s < SEQ; ++s) {
    const float f0 = feats[2 * s + 0];
    const float f1 = feats[2 * s + 1];
    const float comb[6] = { f0, f1, hx[0], hx[1], hx[2], hx[3] };

    float pre[4];
#pragma unroll
    for (int o = 0; o < 4; ++o) {
      float acc = bg[o];
#pragma unroll
      for (int i = 0; i < 6; ++i) acc = fmaf(Wg[o * 6 + i], comb[i], acc);
      pre[o] = acc;
    }

    // --- qgate: 16-amplitude statevector sim ---
    float ax[16], ay[16];
#pragma unroll
    for (int i = 0; i < 16; ++i) { ax[i] = 0.f; ay[i] = 0.f; }
    ax[0] = 1.f;
#pragma unroll
    for (int w = 0; w < 4; ++w)
      rx16(ax, ay, 0.5f * (pre[w] + th[w]), 8 >> w);
    cnot16(ax, ay, 8, 4);  // CNOT(0,1)
    cnot16(ax, ay, 4, 2);  // CNOT(1,2)
    cnot16(ax, ay, 2, 1);  // CNOT(2,3)
    cnot16(ax, ay, 1, 8);  // CNOT(3,0)

    float z[4] = {0.f, 0.f, 0.f, 0.f};
#pragma unroll
    for (int i = 0; i < 16; ++i) {
      const float p = ax[i] * ax[i] + ay[i] * ay[i];
      z[0] += (i & 8) ? -p : p;
      z[1] += (i & 4) ? -p : p;
      z[2] += (i & 2) ? -p : p;
      z[3] += (i & 1) ? -p : p;
    }

    // exchange gate results across lanes 0..3 (wave32 shuffle)
#pragma unroll
    for (int o = 0; o < 4; ++o) {
      const float zf = __shfl(z[o], 0, 32);
      const float zi = __shfl(z[o], 1, 32);
      const float zg = __shfl(z[o], 2, 32);
      const float zo = __shfl(z[o], 3, 32);
      const float fg = sigmoidf(zf);
      const float ig = sigmoidf(zi);
      const float gg = tanhf(zg);
      const float og = sigmoidf(zo);
      cx[o] = fg * cx[o] + ig * gg;
      hx[o] = og * tanhf(cx[o]);
    }

    if (lane == 0) {
      float4 h; h.x = hx[0]; h.y = hx[1]; h.z = hx[2]; h.w = hx[3];
      *reinterpret_cast<float4*>(hs + 4 * s) = h;
    }
  }
}

// ---------------------------------------------------------------------------
// Kernel 3: tag projection via V_WMMA_F32_16X16X4_F32 + row log_softmax
// One wave per 16-row stripe; 8 WMMAs cover all 128 tags; tile in LDS.
// A (16x4 f32): lane l -> M = l&15, K = j + 2*(l>>4)   (float2 per lane)
// B (4x16 f32): lane l -> N = l&15, K = j + 2*(l>>4)   (B = W_tag^T)
// C/D (16x16): vgpr j, lane l -> M = j + 8*(l>>4), N = l&15
// ---------------------------------------------------------------------------
__global__ void __launch_bounds__(32)
tag_kernel(const float* __restrict__ hs, const float* __restrict__ Wtag,
           const float* __restrict__ btag, float* __restrict__ out) {
  __shared__ float tile[16][TAGS + 4];   // padded row stride vs bank conflicts

  const int lane = threadIdx.x;
  const int r0   = blockIdx.x * 16;
  const int lrow = lane & 15;
  const int lhi  = lane >> 4;

  // A fragment: rows r0..r0+15 of hs, K = 2*lhi .. 2*lhi+1
  const v2f a = *reinterpret_cast<const v2f*>(hs + (r0 + lrow) * 4 + 2 * lhi);

#pragma unroll
  for (int t = 0; t < 8; ++t) {
    const int n0 = t * 16;
    // B fragment: B[k][n] = W_tag[n][k]
    const v2f bb = *reinterpret_cast<const v2f*>(Wtag + (n0 + lrow) * 4 + 2 * lhi);
    const float bv = btag[n0 + lrow];        // bias depends only on N
    v8f c;
#pragma unroll
    for (int j = 0; j < 8; ++j) c[j] = bv;

    v8f d = __builtin_amdgcn_wmma_f32_16x16x4_f32(
        /*neg_a=*/false, a, /*neg_b=*/false, bb,
        /*c_mod=*/(short)0, c, /*reuse_a=*/false, /*reuse_b=*/false);

#pragma unroll
    for (int j = 0; j < 8; ++j)
      tile[j + 8 * lhi][n0 + lrow] = d[j];
  }
  __syncthreads();

  // log_softmax over 128 tags per row: 2 lanes per row, 64 cols each
  const int row  = lane & 15;
  const int cbase = lhi * 64;
  float m = -INFINITY;
#pragma unroll 8
  for (int c = 0; c < 64; ++c) m = fmaxf(m, tile[row][cbase + c]);
  m = fmaxf(m, __shfl_xor(m, 16, 32));
  float sum = 0.f;
#pragma unroll 8
  for (int c = 0; c < 64; ++c) sum += __expf(tile[row][cbase + c] - m);
  sum += __shfl_xor(sum, 16, 32);
  const float lse = m + __logf(sum);
#pragma unroll 8
  for (int c = 0; c < 64; ++c) {
    const int col = cbase + c;
    out[(r0 + row) * TAGS + col] = tile[row][col] - lse;
  }
}

// ---------------------------------------------------------------------------
// launch
// ---------------------------------------------------------------------------
extern "C" void kernel_launch(void* const* d_in, const int* in_sizes, int n_in,
                              void* d_out, int out_size, void* d_ws, size_t ws_size,
                              hipStream_t stream) {
  (void)in_sizes; (void)n_in; (void)out_size; (void)ws_size;
  const float* emb      = (const float*)d_in[0];   // [50257, 2]
  const float* W_gates  = (const float*)d_in[1];   // [4, 4, 6]
  const float* b_gates  = (const float*)d_in[2];   // [4, 4]
  const float* qtheta   = (const float*)d_in[3];   // [4, 4]
  const float* W_tag    = (const float*)d_in[4];   // [128, 4]
  const float* b_tag    = (const float*)d_in[5];   // [128]
  const int*   sentence = (const int*)d_in[6];     // [8192]
  float* out = (float*)d_out;                      // [8192, 128]

  float* feats = (float*)d_ws;                     // [8192, 2]
  float* hs    = feats + SEQ * 2;                  // [8192, 4]

  photonic_kernel<<<SEQ / 256, 256, 0, stream>>>(emb, sentence, feats);
  qlstm_kernel<<<1, 32, 0, stream>>>(feats, W_gates, b_gates, qtheta, hs);
  tag_kernel<<<SEQ / 16, 32, 0, stream>>>(hs, W_tag, b_tag, out);
}